// EMCriterion_64836826300503
// MI455X (gfx1250) — compile-verified
//
#include <hip/hip_runtime.h>
#include <math.h>

#define BB 4
#define QQ 1024
#define EE 200
#define HH 128
#define WW 128
#define HWPIX (HH * WW)          // 16384 pixels per image
#define BQ (BB * QQ)             // 4096
#define BE (BB * EE)             // 800
#define CHUNKS 128               // blocks per image for the seg kernel
#define PPB (HWPIX / CHUNKS)     // 128 pixels per block

// ---- CDNA5 TDM availability (compile-safe guards) -------------------------
#if defined(__AMDGCN__) && __has_builtin(__builtin_amdgcn_tensor_load_to_lds) && \
    __has_builtin(__builtin_amdgcn_s_wait_tensorcnt)
#define HAVE_TDM 1
#else
#define HAVE_TDM 0
#endif

#if __has_include(<hip/amd_detail/amd_gfx1250_TDM.h>)
#define TDM_SIX_ARGS 1           // amdgpu-toolchain (therock headers): 6-arg builtin
#else
#define TDM_SIX_ARGS 0           // ROCm 7.2: 5-arg builtin
#endif

typedef __attribute__((ext_vector_type(2))) float v2f;
typedef __attribute__((ext_vector_type(8))) float v8f;
typedef __attribute__((ext_vector_type(4))) unsigned int u32x4;
typedef __attribute__((ext_vector_type(4))) int i32x4;
typedef __attribute__((ext_vector_type(8))) int i32x8;

// Workspace layout (floats):
//  acc[0]=class_sum acc[1]=bce_sum acc[2]=union_cnt acc[3]=nll_sum
//  acc[4]=huber_sum acc[5]=salience_sum
//  [8,808) dice_num   [808,1608) dice_p   [1608,2408) dice_t
#define WS_FLOATS 2408

__device__ __forceinline__ float fast_exp(float x) { return __expf(x); }
__device__ __forceinline__ float fast_sigmoid(float x) {
    return __builtin_amdgcn_rcpf(1.0f + __expf(-x));
}
__device__ __forceinline__ float bce_logits(float x, float t) {
    return fmaxf(x, 0.0f) - x * t + log1pf(fast_exp(-fabsf(x)));
}

// Finish a WMMA ones-matrix reduction chain.
// C layout (16x16 f32): lane L<16: c[j]=C[j][L]; lane L>=16: c[j]=C[j+8][L-16].
// All columns identical (B==ones): total = s(lane0)+s(lane16).
__device__ __forceinline__ float wmma_chain_total(v8f c) {
    float s = c[0] + c[1] + c[2] + c[3] + c[4] + c[5] + c[6] + c[7];
    return __shfl(s, 0, 32) + __shfl(s, 16, 32);
}

#if HAVE_TDM
// Issue one 1-D TDM copy: n elements of (1<<dszLog) bytes, global g -> LDS l.
// D# layout per CDNA5 ISA (group0 128b, group1 256b).
__device__ __forceinline__ void tdm_copy_1d(unsigned long long g, unsigned int l,
                                            unsigned int n, unsigned int dszCode) {
    u32x4 g0;
    g0.x = 1u;                                            // count=1, user mode
    g0.y = l;                                             // lds_addr (bytes)
    g0.z = (unsigned int)g;                               // global_addr[31:0]
    g0.w = (unsigned int)((g >> 32) & 0x01FFFFFFull)      // global_addr[56:32]
         | 0x80000000u;                                   // type=2 ("image")
    i32x8 g1;
    g1[0] = (int)(dszCode << 16);                         // wg_mask=0, data_size
    g1[1] = (int)((n & 0xFFFFu) << 16);                   // tensor_dim0[15:0]
    g1[2] = (int)(((n >> 16) & 0xFFFFu) | (1u << 16));    // td0[31:16] | td1=1
    g1[3] = (int)(n << 16);                               // tile_dim0 (n<=65535)
    g1[4] = 0;                                            // tile_dim1=0 (1-D)
    g1[5] = (int)n;                                       // td0_stride lo
    g1[6] = 0;
    g1[7] = 0;
    i32x4 z4 = {0, 0, 0, 0};
#if TDM_SIX_ARGS
    i32x8 z8 = {0, 0, 0, 0, 0, 0, 0, 0};
    __builtin_amdgcn_tensor_load_to_lds(g0, g1, z4, z4, z8, 0);
#else
    __builtin_amdgcn_tensor_load_to_lds(g0, g1, z4, z4, 0);
#endif
}
#endif

// ---------------------------------------------------------------------------
// Kernel A: class loss, label==0 bulk term (weight 0.1). 16 blocks x 256.
// ---------------------------------------------------------------------------
__global__ void class_base_kernel(const float* __restrict__ logits,
                                  float* __restrict__ acc) {
    __shared__ float red[256];
    int i = blockIdx.x * blockDim.x + threadIdx.x;
    float x = logits[i];
    float v = 0.1f * (fmaxf(x, 0.0f) + log1pf(fast_exp(-fabsf(x))));
    red[threadIdx.x] = v;
    __syncthreads();
    for (int s = 128; s > 0; s >>= 1) {
        if (threadIdx.x < (unsigned)s) red[threadIdx.x] += red[threadIdx.x + s];
        __syncthreads();
    }
    if (threadIdx.x == 0) atomicAdd(&acc[0], red[0]);
}

// ---------------------------------------------------------------------------
// Kernel B: segmentation BCE / union / dice partials. Grid = BB*CHUNKS blocks,
// 256 threads (8 waves). Lane <-> electron e (masked beyond E=200; EXEC stays
// full so WMMA is legal). Per pixel the four rows (pred 4KB, pmask 1KB,
// true 800B, tmask 200B) are staged to LDS by the Tensor Data Mover (wave 0
// issues, TENSORcnt-pipelined 2 pixel sets (4 descriptors each) deep, triple-buffered), then
// gathered from LDS. BCE-sum / union-count accumulate on the matrix unit via
// chained v_wmma_f32_16x16x4_f32 against an all-ones B operand (exact f32).
// ---------------------------------------------------------------------------
#define BUF_STRIDE 6400u         // bytes per LDS staging buffer
#define OFF_PMASK 4096u
#define OFF_TRUE  5120u
#define OFF_TMASK 5920u

__global__ void seg_kernel(const float* __restrict__ pred_seg,
                           const unsigned char* __restrict__ pred_mask,
                           const float* __restrict__ true_seg,
                           const unsigned char* __restrict__ true_mask,
                           const int* __restrict__ midx,
                           const int* __restrict__ tidx,
                           float* __restrict__ acc,
                           float* __restrict__ dnum,
                           float* __restrict__ dp,
                           float* __restrict__ dt) {
    const int b = blockIdx.x / CHUNKS;
    const int chunk = blockIdx.x % CHUNKS;
    const int p0 = chunk * PPB;
    const int e = threadIdx.x;
    const bool act = (e < EE);
    const int qi = act ? midx[b * EE + e] : 0;
    const int ti = act ? tidx[b * EE + e] : 0;

    const v2f ones = {1.0f, 1.0f};
    v8f cb = {};   // BCE-sum matrix accumulator
    v8f cu = {};   // union-count matrix accumulator
    float dn = 0.0f, dpv = 0.0f, dtv = 0.0f;

#if HAVE_TDM
    __shared__ float ldsf[3 * BUF_STRIDE / 4];
    unsigned char* ldsb = (unsigned char*)ldsf;
    const unsigned int ldsBase = (unsigned int)(unsigned long long)(uintptr_t)ldsb;
    const unsigned long long gp  = (unsigned long long)(uintptr_t)pred_seg;
    const unsigned long long gpm = (unsigned long long)(uintptr_t)pred_mask;
    const unsigned long long gt  = (unsigned long long)(uintptr_t)true_seg;
    const unsigned long long gtm = (unsigned long long)(uintptr_t)true_mask;

    auto issue_set = [&](int p) {
        const unsigned long long pix = (unsigned long long)(b * HWPIX + p0 + p);
        const unsigned int base = ldsBase + (unsigned int)(p % 3) * BUF_STRIDE;
        tdm_copy_1d(gp  + pix * (unsigned long long)(QQ * 4), base,             QQ, 2u);
        tdm_copy_1d(gpm + pix * (unsigned long long)QQ,       base + OFF_PMASK, QQ, 0u);
        tdm_copy_1d(gt  + pix * (unsigned long long)(EE * 4), base + OFF_TRUE,  EE, 2u);
        tdm_copy_1d(gtm + pix * (unsigned long long)EE,       base + OFF_TMASK, EE, 0u);
    };

    if (threadIdx.x < 32) {           // wave 0 only: TDM ignores EXEC,
        issue_set(0);                 // gate by wave to avoid duplicate DMAs
        issue_set(1);
    }

    for (int p = 0; p < PPB; ++p) {
        if (threadIdx.x < 32) {
            if (p + 1 < PPB) __builtin_amdgcn_s_wait_tensorcnt(4);  // set p done
            else             __builtin_amdgcn_s_wait_tensorcnt(0);
        }
        __syncthreads();              // publish buffer p; everyone past p-1
        asm volatile("" ::: "memory");
        if ((threadIdx.x < 32) && (p + 2 < PPB)) issue_set(p + 2);

        const unsigned int boff = (unsigned int)(p % 3) * BUF_STRIDE;
        float pv = *(const float*)(ldsb + boff + 4u * (unsigned)qi);
        bool  pm = act && (ldsb[boff + OFF_PMASK + (unsigned)qi] != 0);
        float tv = *(const float*)(ldsb + boff + OFF_TRUE + 4u * (unsigned)ti);
        bool  tm = act && (ldsb[boff + OFF_TMASK + (unsigned)ti] != 0);
#else
    for (int p = 0; p < PPB; ++p) {
        const size_t pix = (size_t)(b * HWPIX + p0 + p);
        const size_t pb = pix * (size_t)QQ;
        const size_t tb = pix * (size_t)EE;
        float pv = pred_seg[pb + qi];
        bool  pm = act && (pred_mask[pb + qi] != 0);
        float tv = true_seg[tb + ti];
        bool  tm = act && (true_mask[tb + ti] != 0);
        if (p + 1 < PPB) {
            __builtin_prefetch(&pred_seg[pb + QQ + qi], 0, 3);
            __builtin_prefetch(&true_seg[tb + EE + ti], 0, 3);
        }
#endif
        float x = pm ? pv : 0.0f;            // pred_vals
        float t = tm ? tv : 0.0f;            // true_vals
        bool  u = pm || tm;                  // union support
        float bc = u ? bce_logits(x, t) : 0.0f;
        float uc = u ? 1.0f : 0.0f;

        v2f ab = {bc, 0.0f};
        v2f au = {uc, 0.0f};
        cb = __builtin_amdgcn_wmma_f32_16x16x4_f32(false, ab, false, ones,
                                                   (short)0, cb, false, false);
        cu = __builtin_amdgcn_wmma_f32_16x16x4_f32(false, au, false, ones,
                                                   (short)0, cu, false, false);

        float ps = pm ? fast_sigmoid(x) : 0.0f;   // sigmoid only on stored
        dn  += ps * t;
        dpv += ps;
        dtv += t;
    }

    float bce_tot = wmma_chain_total(cb);
    float uni_tot = wmma_chain_total(cu);
    if ((threadIdx.x & 31u) == 0u) {
        atomicAdd(&acc[1], bce_tot);
        atomicAdd(&acc[2], uni_tot);
    }
    if (act) {
        atomicAdd(&dnum[b * EE + e], dn);
        atomicAdd(&dp[b * EE + e], dpv);
        atomicAdd(&dt[b * EE + e], dtv);
    }
}

// ---------------------------------------------------------------------------
// Kernel C: per-matched-pair NLL + Huber + class-loss correction. 4 x 256.
// ---------------------------------------------------------------------------
__global__ void matched_kernel(const float* __restrict__ logits,
                               const float* __restrict__ pred_pos,
                               const float* __restrict__ chol,
                               const float* __restrict__ true_pos,
                               const int* __restrict__ midx,
                               const int* __restrict__ tidx,
                               float* __restrict__ acc) {
    __shared__ float rn[256], rh[256], rc[256];
    const int i = blockIdx.x * blockDim.x + threadIdx.x;
    float nll = 0.0f, hub = 0.0f, cls = 0.0f;
    if (i < BE) {
        const int b = i / EE;
        const int pq = midx[i];
        const int tq = tidx[i];
        const int pr = b * QQ + pq;
        const int tr = b * EE + tq;
        float px = pred_pos[pr * 2 + 0], py = pred_pos[pr * 2 + 1];
        float L00 = chol[pr * 4 + 0];
        float L10 = chol[pr * 4 + 2];
        float L11 = chol[pr * 4 + 3];
        float tx = true_pos[tr * 2 + 0], ty = true_pos[tr * 2 + 1];
        float d0 = tx - px, d1 = ty - py;
        float z0 = d0 / L00;
        float z1 = (d1 - L10 * z0) / L11;
        nll = 0.5f * (z0 * z0 + z1 * z1) + logf(L00) + logf(L11)
              + 1.8378770664093453f;  // log(2*pi)
        float a0 = fabsf(d0), a1 = fabsf(d1);
        hub = (a0 < 1.0f ? 0.5f * a0 * a0 : a0 - 0.5f)
            + (a1 < 1.0f ? 0.5f * a1 * a1 : a1 - 0.5f);
        // class correction: (w=1,label=1 term) - (w=0.1,label=0 term)
        float x = logits[pr];
        cls = 0.9f * (fmaxf(x, 0.0f) + log1pf(fast_exp(-fabsf(x)))) - x;
    }
    rn[threadIdx.x] = nll; rh[threadIdx.x] = hub; rc[threadIdx.x] = cls;
    __syncthreads();
    for (int s = 128; s > 0; s >>= 1) {
        if (threadIdx.x < (unsigned)s) {
            rn[threadIdx.x] += rn[threadIdx.x + s];
            rh[threadIdx.x] += rh[threadIdx.x + s];
            rc[threadIdx.x] += rc[threadIdx.x + s];
        }
        __syncthreads();
    }
    if (threadIdx.x == 0) {
        atomicAdd(&acc[3], rn[0]);
        atomicAdd(&acc[4], rh[0]);
        atomicAdd(&acc[0], rc[0]);
    }
}

// ---------------------------------------------------------------------------
// Kernel D: salience focal loss, WMMA ones-reduction chain. 32 x 256.
// ---------------------------------------------------------------------------
__device__ __forceinline__ float focal_term(float x, float t) {
    float ps = fast_sigmoid(x);
    float ce = bce_logits(x, t);
    float pt = ps * t + (1.0f - ps) * (1.0f - t);
    float at = 0.25f * t + 0.75f * (1.0f - t);
    float om = 1.0f - pt;
    return at * ce * om * om;
}

__global__ void salience_kernel(const float* __restrict__ slog,
                                const float* __restrict__ stgt,
                                float* __restrict__ acc) {
    const int N = BB * HWPIX;
    const int stride = gridDim.x * blockDim.x;
    const v2f ones = {1.0f, 1.0f};
    v8f c = {};
    for (int i = blockIdx.x * blockDim.x + threadIdx.x; i < N; i += 2 * stride) {
        float v0 = focal_term(slog[i], stgt[i]);
        int j = i + stride;
        float v1 = (j < N) ? focal_term(slog[j], stgt[j]) : 0.0f;
        v2f a = {v0, v1};
        c = __builtin_amdgcn_wmma_f32_16x16x4_f32(false, a, false, ones,
                                                  (short)0, c, false, false);
    }
    float tot = wmma_chain_total(c);
    if ((threadIdx.x & 31u) == 0u) atomicAdd(&acc[5], tot);
}

// ---------------------------------------------------------------------------
// Kernel E: finalize — dice mean over 800 entries + assemble 7 outputs.
// ---------------------------------------------------------------------------
__global__ void finalize_kernel(const float* __restrict__ acc,
                                const float* __restrict__ dnum,
                                const float* __restrict__ dp,
                                const float* __restrict__ dt,
                                float* __restrict__ out) {
    __shared__ float red[256];
    float s = 0.0f;
    for (int i = threadIdx.x; i < BE; i += 256) {
        float num = 2.0f * dnum[i];
        float den = dp[i] + dt[i];
        s += 1.0f - (num + 1.0f) / (den + 1.0f);
    }
    red[threadIdx.x] = s;
    __syncthreads();
    for (int st = 128; st > 0; st >>= 1) {
        if (threadIdx.x < (unsigned)st) red[threadIdx.x] += red[threadIdx.x + st];
        __syncthreads();
    }
    if (threadIdx.x == 0) {
        float class_loss = acc[0] / (float)BQ;
        float bce_loss   = acc[1] / fmaxf(acc[2], 1.0f);
        float dice_loss  = red[0] / (float)BE;
        float nll_loss   = acc[3] / (float)BE;
        float huber_loss = acc[4] / (float)(2 * BE);
        float sal_loss   = acc[5] / (float)(BB * HWPIX);
        out[0] = class_loss;
        out[1] = bce_loss;
        out[2] = dice_loss;
        out[3] = nll_loss;
        out[4] = huber_loss;
        out[5] = sal_loss;
        out[6] = class_loss + bce_loss + dice_loss + nll_loss + huber_loss + sal_loss;
    }
}

// ---------------------------------------------------------------------------
extern "C" void kernel_launch(void* const* d_in, const int* in_sizes, int n_in,
                              void* d_out, int out_size, void* d_ws, size_t ws_size,
                              hipStream_t stream) {
    (void)in_sizes; (void)n_in; (void)out_size; (void)ws_size;

    const float*         pred_logits = (const float*)d_in[0];
    const float*         pred_seg    = (const float*)d_in[1];
    const unsigned char* pred_mask   = (const unsigned char*)d_in[2]; // bool
    const float*         true_seg    = (const float*)d_in[3];
    const unsigned char* true_mask   = (const unsigned char*)d_in[4]; // bool
    const float*         pred_pos    = (const float*)d_in[5];
    const float*         chol        = (const float*)d_in[6];
    const float*         true_pos    = (const float*)d_in[7];
    const float*         slog        = (const float*)d_in[8];
    const float*         stgt        = (const float*)d_in[9];
    const int*           midx        = (const int*)d_in[10];
    const int*           tidx        = (const int*)d_in[11];

    float* ws   = (float*)d_ws;
    float* acc  = ws;            // 8 accumulator slots
    float* dnum = ws + 8;        // 800
    float* dpw  = ws + 808;      // 800
    float* dtw  = ws + 1608;     // 800
    float* out  = (float*)d_out;

    hipMemsetAsync(d_ws, 0, WS_FLOATS * sizeof(float), stream);

    class_base_kernel<<<BQ / 256, 256, 0, stream>>>(pred_logits, acc);

    seg_kernel<<<BB * CHUNKS, 256, 0, stream>>>(pred_seg, pred_mask,
                                                true_seg, true_mask,
                                                midx, tidx,
                                                acc, dnum, dpw, dtw);

    matched_kernel<<<(BE + 255) / 256, 256, 0, stream>>>(pred_logits, pred_pos,
                                                         chol, true_pos,
                                                         midx, tidx, acc);

    salience_kernel<<<32, 256, 0, stream>>>(slog, stgt, acc);

    finalize_kernel<<<1, 256, 0, stream>>>(acc, dnum, dpw, dtw, out);
}